// GNNModel_37177236914701
// MI455X (gfx1250) — compile-verified
//
#include <hip/hip_runtime.h>

typedef float v2f __attribute__((ext_vector_type(2)));
typedef float v8f __attribute__((ext_vector_type(8)));

#define FAGCN_EPS 0.3f

// ---------------- degree / norm ----------------
__global__ void deg_kernel(const int* __restrict__ ei, float* __restrict__ deg, int E) {
  int e = blockIdx.x * blockDim.x + threadIdx.x;
  if (e < E) atomicAdd(&deg[ei[e]], 1.0f);   // row = ei[0:E]
}

__global__ void dn_kernel(float* dn, int N) {
  int i = blockIdx.x * blockDim.x + threadIdx.x;
  if (i < N) dn[i] = 1.0f / sqrtf(fmaxf(dn[i], 1.0f));
}

// ---------------- h = relu(x @ W_in + b_in), W_in: [128,128] ----------------
// One wave computes one 16x16 output tile; 8 waves/block cover the 8 column tiles.
// Out-of-range M rows are CLAMPED (not zeroed): tile row m only feeds output row m,
// which is masked at the store, so clamped garbage is harmless and loads stay
// branch-free. Full-tile stores take a wave-uniform fast path (no exec churn).
__global__ void gemm_relu_kernel(const float* __restrict__ X, const float* __restrict__ W,
                                 const float* __restrict__ bias, float* __restrict__ H,
                                 int Nrows) {
  int wave = threadIdx.x >> 5;          // column tile 0..7
  int lane = threadIdx.x & 31;
  int tileM = blockIdx.x;
  int mA = tileM * 16 + (lane & 15);    // A row for this lane
  int mclamp = mA < Nrows ? mA : (Nrows - 1);
  int n = wave * 16 + (lane & 15);      // B/D column for this lane
  int khalf = (lane >> 4) * 2;          // K sub-offset per ISA 16x4 layout
  const float* xrow = X + (size_t)mclamp * 128 + khalf;   // 8B aligned (khalf even)
  const float* wcol = W + (size_t)khalf * 128 + n;
  __builtin_prefetch(xrow, 0, 0);
  v8f acc = {};
#pragma unroll 4
  for (int kk = 0; kk < 128; kk += 4) {
    v2f a = *(const v2f*)(xrow + kk);        // A: K=k0,k0+1 in one b64
    v2f b;
    b.x = wcol[(size_t)kk * 128];            // B: W[k0][n]
    b.y = wcol[(size_t)kk * 128 + 128];      //    W[k0+1][n]
    acc = __builtin_amdgcn_wmma_f32_16x16x4_f32(false, a, false, b, (short)0, acc,
                                               false, false);
  }
  int mbase = tileM * 16 + ((lane >> 4) * 8);   // D layout: M = reg + 8*(lane>=16)
  float bn = bias[n];
  if (tileM * 16 + 16 <= Nrows) {               // uniform: whole tile in range
    float* hp = H + (size_t)mbase * 128 + n;
#pragma unroll
    for (int r = 0; r < 8; ++r)
      hp[(size_t)r * 128] = fmaxf(acc[r] + bn, 0.0f);
  } else {
#pragma unroll
    for (int r = 0; r < 8; ++r) {
      int mm = mbase + r;
      if (mm < Nrows) H[(size_t)mm * 128 + n] = fmaxf(acc[r] + bn, 0.0f);
    }
  }
}

// ---------------- per-node gate vectors: ga = h.Wg[:128], gb = h.Wg[128:] ----------------
__global__ void gate_kernel(const float* __restrict__ H, const float* __restrict__ Wg,
                            float* __restrict__ ga, float* __restrict__ gb, int Nrows) {
  int wave = threadIdx.x >> 5;
  int lane = threadIdx.x & 31;
  int node = blockIdx.x * 8 + wave;
  if (node >= Nrows) return;
  const float* hrow = H + (size_t)node * 128;
  float sa = 0.0f, sb = 0.0f;
#pragma unroll
  for (int j = 0; j < 4; ++j) {
    int f = lane + j * 32;
    float hv = hrow[f];
    sa += hv * Wg[f];
    sb += hv * Wg[128 + f];
  }
  for (int off = 16; off > 0; off >>= 1) {
    sa += __shfl_down(sa, off, 32);
    sb += __shfl_down(sb, off, 32);
  }
  if (lane == 0) { ga[node] = sa; gb[node] = sb; }
}

// ---------------- per-edge gate * degree norm ----------------
__global__ void norm_kernel(const int* __restrict__ ei, const float* __restrict__ ga,
                            const float* __restrict__ gb, const float* __restrict__ dn,
                            const float* __restrict__ bg, int layer,
                            float* __restrict__ nrm, int E) {
  int e = blockIdx.x * blockDim.x + threadIdx.x;
  if (e >= E) return;
  int r = ei[e], c = ei[E + e];
  float g = tanhf(ga[r] + gb[c] + bg[layer]);
  nrm[e] = g * dn[r] * dn[c];
}

// ---------------- scatter-add: agg[col] += nrm[e] * h[row] ----------------
// 32 threads per edge, 4 features each: b128 gather of h[row], 4x atomic f32 add.
__global__ void scatter_kernel(const int* __restrict__ ei, const float* __restrict__ nrm,
                               const float* __restrict__ H, float* __restrict__ agg, int E) {
  long long idx = (long long)blockIdx.x * blockDim.x + threadIdx.x;
  if (idx >= (long long)E * 32) return;
  int e = (int)(idx >> 5);
  int f = (int)(idx & 31) * 4;
  int r = ei[e], c = ei[E + e];
  float s = nrm[e];
  float4 hv = *(const float4*)(H + (size_t)r * 128 + f);
  float* dst = agg + (size_t)c * 128 + f;
  atomicAdd(dst + 0, s * hv.x);
  atomicAdd(dst + 1, s * hv.y);
  atomicAdd(dst + 2, s * hv.z);
  atomicAdd(dst + 3, s * hv.w);
}

// ---------------- h = eps*raw + agg (in place into agg), float4 streams ----------------
__global__ void residual_kernel(const float4* __restrict__ raw, float4* __restrict__ agg,
                                long long n4) {
  long long i = (long long)blockIdx.x * blockDim.x + threadIdx.x;
  if (i < n4) {
    float4 r = raw[i];
    float4 a = agg[i];
    a.x = FAGCN_EPS * r.x + a.x;
    a.y = FAGCN_EPS * r.y + a.y;
    a.z = FAGCN_EPS * r.z + a.z;
    a.w = FAGCN_EPS * r.w + a.w;
    agg[i] = a;
  }
}

// ---------------- out = h @ W_out + b_out, W_out: [128,16] ----------------
__global__ void gemm_out_kernel(const float* __restrict__ H, const float* __restrict__ W,
                                const float* __restrict__ bias, float* __restrict__ out,
                                int Nrows) {
  int wave = threadIdx.x >> 5;
  int lane = threadIdx.x & 31;
  int tileM = blockIdx.x * 8 + wave;
  int tiles = (Nrows + 15) >> 4;
  if (tileM >= tiles) return;               // uniform per wave
  int mA = tileM * 16 + (lane & 15);
  int mclamp = mA < Nrows ? mA : (Nrows - 1);
  int n = lane & 15;                        // OUT == 16 -> single column tile
  int khalf = (lane >> 4) * 2;
  const float* hrow = H + (size_t)mclamp * 128 + khalf;
  const float* wcol = W + (size_t)khalf * 16 + n;
  __builtin_prefetch(hrow, 0, 0);
  v8f acc = {};
#pragma unroll 4
  for (int kk = 0; kk < 128; kk += 4) {
    v2f a = *(const v2f*)(hrow + kk);
    v2f b;
    b.x = wcol[(size_t)kk * 16];
    b.y = wcol[(size_t)kk * 16 + 16];
    acc = __builtin_amdgcn_wmma_f32_16x16x4_f32(false, a, false, b, (short)0, acc,
                                               false, false);
  }
  int mbase = tileM * 16 + ((lane >> 4) * 8);
  float bn = bias[n];
  if (tileM * 16 + 16 <= Nrows) {               // uniform fast path
    float* op = out + (size_t)mbase * 16 + n;
#pragma unroll
    for (int r = 0; r < 8; ++r)
      op[(size_t)r * 16] = acc[r] + bn;
  } else {
#pragma unroll
    for (int r = 0; r < 8; ++r) {
      int mm = mbase + r;
      if (mm < Nrows) out[(size_t)mm * 16 + n] = acc[r] + bn;
    }
  }
}

// ---------------- row-wise log_softmax over 16 classes, in place ----------------
__global__ void logsoftmax_kernel(float4* out, int N) {
  int i = blockIdx.x * blockDim.x + threadIdx.x;
  if (i >= N) return;
  float4* row = out + (size_t)i * 4;       // 16 floats = 4x float4
  float4 v[4];
#pragma unroll
  for (int q = 0; q < 4; ++q) v[q] = row[q];
  float m = v[0].x;
#pragma unroll
  for (int q = 0; q < 4; ++q) {
    m = fmaxf(m, v[q].x); m = fmaxf(m, v[q].y);
    m = fmaxf(m, v[q].z); m = fmaxf(m, v[q].w);
  }
  float s = 0.0f;
#pragma unroll
  for (int q = 0; q < 4; ++q) {
    s += expf(v[q].x - m); s += expf(v[q].y - m);
    s += expf(v[q].z - m); s += expf(v[q].w - m);
  }
  float d = m + logf(s);
#pragma unroll
  for (int q = 0; q < 4; ++q) {
    v[q].x -= d; v[q].y -= d; v[q].z -= d; v[q].w -= d;
    row[q] = v[q];
  }
}

extern "C" void kernel_launch(void* const* d_in, const int* in_sizes, int n_in,
                              void* d_out, int out_size, void* d_ws, size_t ws_size,
                              hipStream_t stream) {
  const float* x     = (const float*)d_in[0];
  const int*   ei    = (const int*)d_in[1];    // [2,E]: rows then cols
  const float* W_in  = (const float*)d_in[2];
  const float* b_in  = (const float*)d_in[3];
  const float* Wg    = (const float*)d_in[4];  // [L, 2*128]
  const float* bg    = (const float*)d_in[5];  // [L]
  const float* W_out = (const float*)d_in[6];
  const float* b_out = (const float*)d_in[7];
  float* out = (float*)d_out;

  const int Hdim = 128;
  int N = in_sizes[0] / Hdim;
  int E = in_sizes[1] / 2;
  int L = in_sizes[5];

  // workspace layout (floats)
  float* ws   = (float*)d_ws;
  float* dn   = ws;                          // N
  float* ga   = dn + N;                      // N
  float* gb   = ga + N;                      // N
  float* nrm  = gb + N;                      // E
  float* raw  = nrm + E;                     // N*128 (initial h, kept for residual)
  float* bufA = raw + (size_t)N * Hdim;      // N*128
  float* bufB = bufA + (size_t)N * Hdim;     // N*128

  // degree norm
  hipMemsetAsync(dn, 0, (size_t)N * sizeof(float), stream);
  deg_kernel<<<(E + 255) / 256, 256, 0, stream>>>(ei, dn, E);
  dn_kernel<<<(N + 255) / 256, 256, 0, stream>>>(dn, N);

  // h = relu(x @ W_in + b_in)
  gemm_relu_kernel<<<(N + 15) / 16, 256, 0, stream>>>(x, W_in, b_in, raw, N);

  float* h = raw;
  float* agg = bufA;
  for (int i = 0; i < L; ++i) {
    gate_kernel<<<(N + 7) / 8, 256, 0, stream>>>(h, Wg + (size_t)i * 2 * Hdim, ga, gb, N);
    norm_kernel<<<(E + 255) / 256, 256, 0, stream>>>(ei, ga, gb, dn, bg, i, nrm, E);
    hipMemsetAsync(agg, 0, (size_t)N * Hdim * sizeof(float), stream);
    long long tot = (long long)E * 32;   // 4 features per thread
    scatter_kernel<<<(unsigned)((tot + 255) / 256), 256, 0, stream>>>(ei, nrm, h, agg, E);
    long long n4 = (long long)N * Hdim / 4;
    residual_kernel<<<(unsigned)((n4 + 255) / 256), 256, 0, stream>>>(
        (const float4*)raw, (float4*)agg, n4);
    h = agg;
    agg = (h == bufA) ? bufB : bufA;
  }

  int tiles = (N + 15) / 16;
  gemm_out_kernel<<<(tiles + 7) / 8, 256, 0, stream>>>(h, W_out, b_out, out, N);
  logsoftmax_kernel<<<(N + 255) / 256, 256, 0, stream>>>((float4*)out, N);
}